// computeMaskedOutput_47382079209766
// MI455X (gfx1250) — compile-verified
//
#include <hip/hip_runtime.h>
#include <stdint.h>

// Problem geometry (from reference): B=128, H=W=28 -> HW=784, C=512.
#define HW_     784
#define C_      512
#define CTILE   32          // channels per block
#define NTHR    256         // 8 waves of 32
#define NWAVE   8
#define NITER   ((HW_ * (CTILE / 2)) / NTHR)   // 12544 chunks / 256 thr = 49

// Fused kernel: one block handles (b, 32-channel tile).
//  Phase 1: async-copy x[b,:,:,c0:c0+32] (98 KB) into LDS  (ASYNCcnt path)
//  Phase 2: per-channel argmax over 784 spatial positions (first-max ties)
//  Phase 3: masked = relu(x * t_p[argmax_row, s]); NT-store masked, x, templates
__global__ __launch_bounds__(NTHR) void masked_template_fused(
    const float* __restrict__ x,     // [B, HW, C]
    const float* __restrict__ t_p,   // [HW, HW] (row = h*W+w of argmax)
    float* __restrict__ out,         // [3, B, HW, C] concatenated
    size_t N)                        // B*HW*C
{
    extern __shared__ float xs[];    // HW_ * CTILE floats = 100,352 B (dynamic)
    __shared__ float rvals[NTHR];
    __shared__ int   ridx[NTHR];
    __shared__ int   brow[CTILE];

    const int tid  = threadIdx.x;
    const int b    = blockIdx.x >> 4;           // 16 channel-tiles per batch
    const int c0   = (blockIdx.x & 15) * CTILE;

    const float* gb = x + ((size_t)b * HW_) * C_ + c0;

    // LDS byte address of xs: flat-address low 32 bits == AS3 (LDS) offset.
    const uint32_t lds_base = (uint32_t)(uintptr_t)(void*)xs;

    // ---- Phase 1: async global -> LDS copy, affine addresses.
    // Thread covers chunk (tid>>4 + 16*iter) of 784 spatial rows; 8 B per chunk.
    {
        const int row0 = tid >> 4;          // starting spatial row (0..15)
        const int j    = tid & 15;          // 8-byte chunk within 32-ch row
        const float* gp    = gb + (size_t)row0 * C_ + j * 2;
        uint32_t     laddr = lds_base + (uint32_t)(row0 * (CTILE * 4) + j * 8);
        #pragma unroll 1
        for (int it = 0; it < NITER; ++it) {
            asm volatile("global_load_async_to_lds_b64 %0, %1, off"
                         :: "v"(laddr), "v"(gp) : "memory");
            gp    += 16 * C_;               // +32768 bytes
            laddr += 16 * (CTILE * 4);      // +2048 bytes
        }
    }
    asm volatile("s_wait_asynccnt 0x0" ::: "memory");
    __syncthreads();

    // ---- Phase 2: argmax over 784 spatial positions, lane = channel.
    const int wave = tid >> 5;
    const int lane = tid & 31;
    float best = -__builtin_inff();
    int   bidx = HW_;                    // larger than any real index
    for (int s = wave; s < HW_; s += NWAVE) {     // ascending: keeps first max
        const float v = xs[s * CTILE + lane];     // stride-32 dwords: conflict-free
        if (v > best) { best = v; bidx = s; }
    }
    rvals[tid] = best;
    ridx[tid]  = bidx;
    __syncthreads();

    if (tid < CTILE) {                   // one thread per channel combines 8 partials
        float bv = rvals[tid];
        int   bi = ridx[tid];
        for (int w = 1; w < NWAVE; ++w) {
            const float v = rvals[w * 32 + tid];
            const int   i = ridx[w * 32 + tid];
            if (v > bv || (v == bv && i < bi)) { bv = v; bi = i; }  // first-max tie rule
        }
        brow[tid] = bi;                  // flat spatial index = h*W+w
    }
    __syncthreads();

    // ---- Phase 3: apply template; streaming (non-temporal) output stores.
    const float* trow = t_p + (size_t)brow[lane] * HW_;   // 3136-B row, L2-resident
    float* ob0 = out + ((size_t)b * HW_) * C_ + c0 + lane;   // masked
    float* ob1 = ob0 + N;                                    // x passthrough
    float* ob2 = ob1 + N;                                    // templates
    for (int s = wave; s < HW_; s += NWAVE) {
        const float xv = xs[s * CTILE + lane];
        const float tv = trow[s];
        const float m  = fmaxf(xv * tv, 0.0f);
        const size_t o = (size_t)s * C_;
        __builtin_nontemporal_store(m,  ob0 + o);
        __builtin_nontemporal_store(xv, ob1 + o);
        __builtin_nontemporal_store(tv, ob2 + o);
    }
}

extern "C" void kernel_launch(void* const* d_in, const int* in_sizes, int n_in,
                              void* d_out, int out_size, void* d_ws, size_t ws_size,
                              hipStream_t stream) {
    const float* x   = (const float*)d_in[0];   // [B,H,W,C] f32
    const float* t_p = (const float*)d_in[1];   // [H,W,H,W] f32
    float* out = (float*)d_out;                 // 3x [B,H,W,C] f32 concatenated

    const int    Btot = in_sizes[0] / (HW_ * C_);        // = 128
    const size_t N    = (size_t)Btot * HW_ * C_;
    const int    nblk = Btot * (C_ / CTILE);             // 128 * 16 = 2048

    const size_t shmem = (size_t)HW_ * CTILE * sizeof(float);  // 100,352 B
    masked_template_fused<<<nblk, NTHR, shmem, stream>>>(x, t_p, out, N);
}